// KernalAnsatz_65481071399427
// MI455X (gfx1250) — compile-verified
//
#include <hip/hip_runtime.h>
#include <stdint.h>

typedef float v2f __attribute__((ext_vector_type(2)));
typedef float v8f __attribute__((ext_vector_type(8)));

#define GAMMA 1.0f
#define NN 8192
#define MM 8192
#define DD 256

#define BN 128        // output tile rows (x) per block
#define BM 64         // output tile cols (y) per block
#define KC 32         // K chunk staged in LDS per pipeline stage
#define LDSS 36       // padded LDS row stride (floats): 36 mod 64 -> conflict-free b64 reads
#define NCHUNK (DD / KC)   // 8 pipeline stages

// ---------------------------------------------------------------------------
// Kernel 1: row norms ||x_i||^2 -> ws[0..N), ||y_j||^2 -> ws[N..N+M)
// ---------------------------------------------------------------------------
__global__ __launch_bounds__(256) void rbf_row_norms(
    const float* __restrict__ X, const float* __restrict__ Y,
    float* __restrict__ x2, float* __restrict__ y2)
{
    int t = blockIdx.x * blockDim.x + threadIdx.x;
    const float* src;
    float* dst;
    int row;
    if (t < NN) { src = X; dst = x2; row = t; }
    else        { src = Y; dst = y2; row = t - NN; }
    const float4* p = (const float4*)(src + (size_t)row * DD);
    float s = 0.f;
#pragma unroll 8
    for (int i = 0; i < DD / 4; ++i) {
        float4 v = p[i];
        s += v.x * v.x + v.y * v.y + v.z * v.z + v.w * v.w;
    }
    dst[row] = s;
}

// ---------------------------------------------------------------------------
// Kernel 2: double-buffered async-copy WMMA GEMM (X * Y^T) + fused RBF.
// Block = 256 threads = 8 wave32. Block tile 128x64, wave tile 32x32
// (2x2 of 16x16 f32 accumulators). K pipelined in chunks of 32 through
// two LDS buffers filled with GLOBAL_LOAD_ASYNC_TO_LDS_B128 (ASYNCcnt).
// ---------------------------------------------------------------------------
__global__ __launch_bounds__(256) void rbf_wmma_kernel(
    const float* __restrict__ X, const float* __restrict__ Y,
    const float* __restrict__ x2, const float* __restrict__ y2,
    float* __restrict__ out)
{
    __shared__ float sX[2][BN * LDSS];
    __shared__ float sY[2][BM * LDSS];

    const int tid   = threadIdx.x;
    const int lane  = tid & 31;
    const int wave  = tid >> 5;
    const int waveN = (wave & 3) * 32;   // 4 waves tile the 128 n-rows
    const int waveM = (wave >> 2) * 32;  // 2 waves tile the 64 m-cols

    const int blockN = blockIdx.y * BN;
    const int blockM = blockIdx.x * BM;

    // LDS aperture offsets of the staging buffers (low 32 bits of flat ptr)
    const uint32_t sx_base0 = (uint32_t)(size_t)&sX[0][0];
    const uint32_t sx_base1 = (uint32_t)(size_t)&sX[1][0];
    const uint32_t sy_base0 = (uint32_t)(size_t)&sY[0][0];
    const uint32_t sy_base1 = (uint32_t)(size_t)&sY[1][0];

    const uint64_t xbase = (uint64_t)X;
    const uint64_t ybase = (uint64_t)Y;

    v8f acc00 = {};
    v8f acc01 = {};
    v8f acc10 = {};
    v8f acc11 = {};

    // f32 16x4 A layout: lanes 0-15 hold {K0,K1}, lanes 16-31 hold {K2,K3},
    // row M = lane % 16. B (4x16) mirrors this; since B = Y^T the same LDS
    // indexing serves both operands.
    const int kk_half = (lane >> 4) << 1;  // 0 or 2
    const int lrow    = lane & 15;

    // Issue async HBM -> LDS copies for K-chunk `c` into buffer `buf`.
    auto issue_chunk = [&](int c, int buf) {
        const uint32_t k0 = (uint32_t)c * KC;
        const uint32_t sxb = buf ? sx_base1 : sx_base0;
        const uint32_t syb = buf ? sy_base1 : sy_base0;
        // X: 128 x 32 floats = 1024 float4 -> 4 per thread
#pragma unroll
        for (int i = 0; i < (BN * KC) / (4 * 256); ++i) {
            uint32_t q    = (uint32_t)tid + i * 256u;
            uint32_t r    = q >> 3;          // 8 float4 per row
            uint32_t c4   = q & 7u;
            uint32_t goff = ((uint32_t)(blockN + (int)r) * DD + k0 + c4 * 4u) * 4u;
            uint32_t loff = sxb + (r * LDSS + c4 * 4u) * 4u;
            asm volatile("global_load_async_to_lds_b128 %0, %1, %2"
                         :: "v"(loff), "v"(goff), "s"(xbase) : "memory");
        }
        // Y: 64 x 32 floats = 512 float4 -> 2 per thread
#pragma unroll
        for (int i = 0; i < (BM * KC) / (4 * 256); ++i) {
            uint32_t q    = (uint32_t)tid + i * 256u;
            uint32_t r    = q >> 3;
            uint32_t c4   = q & 7u;
            uint32_t goff = ((uint32_t)(blockM + (int)r) * DD + k0 + c4 * 4u) * 4u;
            uint32_t loff = syb + (r * LDSS + c4 * 4u) * 4u;
            asm volatile("global_load_async_to_lds_b128 %0, %1, %2"
                         :: "v"(loff), "v"(goff), "s"(ybase) : "memory");
        }
    };

    // Consume one staged K-chunk with 32 f32 WMMAs.
    auto compute_chunk = [&](int buf) {
        const float* bX = &sX[buf][0];
        const float* bY = &sY[buf][0];
#pragma unroll
        for (int k = 0; k < KC; k += 4) {
            int kk = k + kk_half;
            v2f a0 = *(const v2f*)(bX + (waveN + lrow)      * LDSS + kk);
            v2f a1 = *(const v2f*)(bX + (waveN + 16 + lrow) * LDSS + kk);
            v2f b0 = *(const v2f*)(bY + (waveM + lrow)      * LDSS + kk);
            v2f b1 = *(const v2f*)(bY + (waveM + 16 + lrow) * LDSS + kk);
            acc00 = __builtin_amdgcn_wmma_f32_16x16x4_f32(false, a0, false, b0,
                                                          (short)0, acc00, false, false);
            acc01 = __builtin_amdgcn_wmma_f32_16x16x4_f32(false, a0, false, b1,
                                                          (short)0, acc01, false, false);
            acc10 = __builtin_amdgcn_wmma_f32_16x16x4_f32(false, a1, false, b0,
                                                          (short)0, acc10, false, false);
            acc11 = __builtin_amdgcn_wmma_f32_16x16x4_f32(false, a1, false, b1,
                                                          (short)0, acc11, false, false);
        }
    };

    // ---- software pipeline over 8 K-chunks, double-buffered LDS ----------
    issue_chunk(0, 0);
#pragma unroll
    for (int c = 0; c < NCHUNK; ++c) {
        // my async copies for chunk c are done; barrier publishes everyone's
        asm volatile("s_wait_asynccnt 0x0" ::: "memory");
        __syncthreads();
        // After this barrier every wave finished computing chunk c-1, so the
        // buffer targeted by chunk c+1 is free to overwrite.
        if (c + 1 < NCHUNK) issue_chunk(c + 1, (c + 1) & 1);
        compute_chunk(c & 1);
    }

    // ---- fused RBF epilogue ----------------------------------------------
    // C/D 16x16 f32 layout: VGPR r holds row M=r (lanes 0-15) / M=r+8
    // (lanes 16-31), column N = lane % 16.
    const int col   = lane & 15;
    const int rbase = (lane >> 4) * 8;

    const int   m0  = blockM + waveM + col;
    const float y2a = y2[m0];
    const float y2b = y2[m0 + 16];

#pragma unroll
    for (int r = 0; r < 8; ++r) {
        const int   n0 = blockN + waveN + rbase + r;
        const int   n1 = n0 + 16;
        const float xa = x2[n0];
        const float xb = x2[n1];
        float d2;

        d2 = fmaxf(xa + y2a - 2.0f * acc00[r], 0.0f);
        out[(size_t)n0 * MM + m0]      = __expf(-GAMMA * d2);
        d2 = fmaxf(xa + y2b - 2.0f * acc01[r], 0.0f);
        out[(size_t)n0 * MM + m0 + 16] = __expf(-GAMMA * d2);
        d2 = fmaxf(xb + y2a - 2.0f * acc10[r], 0.0f);
        out[(size_t)n1 * MM + m0]      = __expf(-GAMMA * d2);
        d2 = fmaxf(xb + y2b - 2.0f * acc11[r], 0.0f);
        out[(size_t)n1 * MM + m0 + 16] = __expf(-GAMMA * d2);
    }
}

// ---------------------------------------------------------------------------
extern "C" void kernel_launch(void* const* d_in, const int* in_sizes, int n_in,
                              void* d_out, int out_size, void* d_ws, size_t ws_size,
                              hipStream_t stream) {
    (void)in_sizes; (void)n_in; (void)out_size; (void)ws_size;

    const float* X = (const float*)d_in[0];
    const float* Y = (const float*)d_in[1];
    float* out = (float*)d_out;

    float* x2 = (float*)d_ws;        // N floats
    float* y2 = x2 + NN;             // M floats

    rbf_row_norms<<<(NN + MM) / 256, 256, 0, stream>>>(X, Y, x2, y2);

    dim3 grid(MM / BM, NN / BN);     // (128, 64)
    rbf_wmma_kernel<<<grid, 256, 0, stream>>>(X, Y, x2, y2, out);
}